// NMS2d_22857815949340
// MI455X (gfx1250) — compile-verified
//
#include <hip/hip_runtime.h>

// NMS2d: out = x where (x - maxpool3x3_same(x) + 1e-5) > 0 else 0
// Input: (32, 1, 1024, 1024) fp32 NCHW. Pure HBM-bound stencil:
// ~268 MB traffic -> ~11.5 us floor at 23.3 TB/s.
//
// Branch-free formulation: for max-pooling, -inf SAME-padding is identical to
// index clamping (the edge element is already in the window), so all halo
// loads are unconditional -- no exec-mask branches, maximal load clustering.

#define IMG_H 1024
#define IMG_W 1024
#define ROWS_PER_BLOCK 4
#define EPSF 1e-5f

typedef __attribute__((ext_vector_type(4))) float v4f;

__device__ __forceinline__ v4f vmax4(v4f a, v4f b) {
  v4f r;
  r.x = fmaxf(a.x, b.x);
  r.y = fmaxf(a.y, b.y);
  r.z = fmaxf(a.z, b.z);
  r.w = fmaxf(a.w, b.w);
  return r;
}

__global__ __launch_bounds__(256, 1) void nms2d_maxpool_kernel(
    const float* __restrict__ x, float* __restrict__ out) {
  const int tid = threadIdx.x;                 // 0..255, wave32 -> 8 waves
  const int x0  = tid << 2;                    // 4 contiguous outputs/thread
  // Clamped halo columns (== -inf padding for a max window)
  const int xl  = (x0 == 0) ? 0 : x0 - 1;
  const int xr  = (x0 + 4 > IMG_W - 1) ? (IMG_W - 1) : (x0 + 4);
  const int y0  = blockIdx.x * ROWS_PER_BLOCK;
  const size_t img_off = (size_t)blockIdx.y * (IMG_H * IMG_W);
  const float* __restrict__ img = x + img_off;
  float* __restrict__ o = out + img_off;

  v4f h[ROWS_PER_BLOCK + 2];   // horizontal 3-max per loaded row
  v4f c[ROWS_PER_BLOCK];       // raw center values (needed for the mask)

#pragma unroll
  for (int k = 0; k < ROWS_PER_BLOCK + 2; ++k) {
    int ry = y0 - 1 + k;                       // clamped row (== -inf pad)
    ry = (ry < 0) ? 0 : ry;
    ry = (ry > IMG_H - 1) ? (IMG_H - 1) : ry;  // scalar min/max, uniform
    const float* row = img + (size_t)ry * IMG_W;
    v4f   cv = *reinterpret_cast<const v4f*>(row + x0);  // global_load_b128
    float l  = row[xl];                        // same cacheline as neighbor
    float r  = row[xr];
    if (k >= 1 && k <= ROWS_PER_BLOCK) c[k - 1] = cv;
    v4f hh;
    hh.x = fmaxf(fmaxf(l,    cv.x), cv.y);
    hh.y = fmaxf(fmaxf(cv.x, cv.y), cv.z);
    hh.z = fmaxf(fmaxf(cv.y, cv.z), cv.w);
    hh.w = fmaxf(fmaxf(cv.z, cv.w), r);
    h[k] = hh;
  }

#pragma unroll
  for (int k = 0; k < ROWS_PER_BLOCK; ++k) {
    v4f mp = vmax4(vmax4(h[k], h[k + 1]), h[k + 2]);
    v4f cv = c[k];
    v4f r;
    r.x = (cv.x - mp.x + EPSF > 0.0f) ? cv.x : 0.0f;
    r.y = (cv.y - mp.y + EPSF > 0.0f) ? cv.y : 0.0f;
    r.z = (cv.z - mp.z + EPSF > 0.0f) ? cv.z : 0.0f;
    r.w = (cv.w - mp.w + EPSF > 0.0f) ? cv.w : 0.0f;
    // Output never re-read: NT store keeps the 192MB L2 free for input rows.
    v4f* dst = reinterpret_cast<v4f*>(o + (size_t)(y0 + k) * IMG_W + x0);
    __builtin_nontemporal_store(r, dst);
  }
}

extern "C" void kernel_launch(void* const* d_in, const int* in_sizes, int n_in,
                              void* d_out, int out_size, void* d_ws, size_t ws_size,
                              hipStream_t stream) {
  const float* x = (const float*)d_in[0];
  float* out = (float*)d_out;
  const int n_images = in_sizes[0] / (IMG_H * IMG_W);   // 32 (N*C)
  dim3 grid(IMG_H / ROWS_PER_BLOCK, n_images, 1);       // 256 x 32 blocks
  nms2d_maxpool_kernel<<<grid, dim3(256, 1, 1), 0, stream>>>(x, out);
}